// TwoLayerGCN_31482110279899
// MI455X (gfx1250) — compile-verified
//
#include <hip/hip_runtime.h>
#include <hip/hip_bf16.h>

typedef __attribute__((ext_vector_type(2))) float v2f;
typedef __attribute__((ext_vector_type(8))) float v8f;

#define IN_C 128
#define LDS_STRIDE 132   // 128 + 4 pad: lane row-stride -> 4-bank stride, conflict-free b64 reads

// ---------------- degree / dinv ----------------
__global__ void gcn_deg_init(float* __restrict__ deg, int n) {
    int i = blockIdx.x * blockDim.x + threadIdx.x;
    if (i < n) deg[i] = 1.0f;                       // self-loop weight
}

__global__ void gcn_deg_accum(const int* __restrict__ dst, const float* __restrict__ ew,
                              float* __restrict__ deg, int n_edges) {
    int e = blockIdx.x * blockDim.x + threadIdx.x;
    if (e < n_edges) atomicAdd(&deg[dst[e]], ew[e]);
}

__global__ void gcn_deg_to_dinv(float* __restrict__ deg, int n) {
    int i = blockIdx.x * blockDim.x + threadIdx.x;
    if (i < n) {
        float d = deg[i];
        deg[i] = (d > 0.0f) ? rsqrtf(d) : 0.0f;
    }
}

// ---------------- WMMA GEMM: H = X @ W ; also AggInit = H*dinv^2 + bias ----------------
// grid.x = ceil(n/16); blockDim.x = 2*cout (8 waves for cout=128, 4 for cout=64).
// Wave w computes the 16x16 tile at columns [16w, 16w+16).
__global__ void gcn_gemm_wmma(const float* __restrict__ X,     // [n, 128]
                              const float* __restrict__ W,     // [128, cout]
                              const float* __restrict__ bias,  // [cout]
                              const float* __restrict__ dinv,  // [n]
                              float* __restrict__ Hraw,        // [n, cout]
                              float* __restrict__ AggInit,     // [n, cout]
                              int n, int cout) {
    __shared__ float lds_a[16 * LDS_STRIDE];
    const int r0  = blockIdx.x * 16;
    const int tid = threadIdx.x;
    const int nthreads = blockDim.x;

    // Async stage of the 16x128 A tile into LDS: direct global->LDS DMA
    // (GLOBAL_LOAD_ASYNC_TO_LDS_B128, ASYNCcnt), no VGPR round trip.
    for (int c = tid; c < 16 * 32; c += nthreads) {
        int row = c >> 5;
        int off = (c & 31) * 4;
        int grow = r0 + row;
        if (grow >= n) grow = n - 1;                 // clamp (dup-load) for tail blocks
        const float* gsrc = X + (size_t)grow * IN_C + off;
        // Low 32 bits of a generic shared-memory pointer == LDS byte offset
        // (LDS aperture tag lives in addr[63:32]).
        unsigned ldsoff = (unsigned)(uintptr_t)(lds_a + row * LDS_STRIDE + off);
        asm volatile("global_load_async_to_lds_b128 %0, %1, off"
                     :: "v"(ldsoff), "v"(gsrc) : "memory");
    }
    asm volatile("s_wait_asynccnt 0x0" ::: "memory");
    __syncthreads();

    const int wave  = tid >> 5;
    const int lane  = tid & 31;
    const int lhalf = lane >> 4;   // 0: lanes 0-15, 1: lanes 16-31
    const int lmod  = lane & 15;
    const int c0    = wave * 16;
    const int col   = c0 + lmod;

    v8f acc = {};
    for (int k = 0; k < IN_C; k += 4) {
        // A fragment (16x4 f32): lane half selects K pair {k,k+1} vs {k+2,k+3}
        v2f a = *(const v2f*)(lds_a + lmod * LDS_STRIDE + k + 2 * lhalf);
        // B fragment (4x16 f32): same K-pair split; coalesced across lmod
        int kb = k + 2 * lhalf;
        v2f b;
        b.x = W[(size_t)kb * cout + col];
        b.y = W[(size_t)(kb + 1) * cout + col];
        acc = __builtin_amdgcn_wmma_f32_16x16x4_f32(
            /*neg_a=*/false, a, /*neg_b=*/false, b,
            /*c_mod=*/(short)0, acc, /*reuse_a=*/false, /*reuse_b=*/false);
    }

    // D layout: lane holds rows (v + 8*lhalf), column = c0 + lmod.
    const float bcol = bias[col];
    if (r0 + 16 <= n) {
        // Uniform fast path: no per-element exec masking.
        #pragma unroll
        for (int v = 0; v < 8; ++v) {
            int row = r0 + v + 8 * lhalf;
            float val = acc[v];
            float di  = dinv[row];
            Hraw[(size_t)row * cout + col]    = val;
            AggInit[(size_t)row * cout + col] = val * di * di + bcol;
        }
    } else {
        for (int v = 0; v < 8; ++v) {
            int row = r0 + v + 8 * lhalf;
            if (row < n) {
                float val = acc[v];
                float di  = dinv[row];
                Hraw[(size_t)row * cout + col]    = val;
                AggInit[(size_t)row * cout + col] = val * di * di + bcol;
            }
        }
    }
}

// ---------------- edge scatter, 128 channels: wave per edge, float4/lane ----------------
__global__ void gcn_edge_scatter128(const int* __restrict__ src, const int* __restrict__ dst,
                                    const float* __restrict__ ew, const float* __restrict__ dinv,
                                    const float* __restrict__ H, float* __restrict__ Agg,
                                    int n_edges) {
    int e = blockIdx.x * (blockDim.x >> 5) + (threadIdx.x >> 5);
    if (e >= n_edges) return;
    int lane = threadIdx.x & 31;
    int s = src[e], d = dst[e];
    float norm = dinv[s] * ew[e] * dinv[d];          // wave-uniform
    const float4 hv = *(const float4*)(H + (size_t)s * 128 + lane * 4);
    float* base = Agg + (size_t)d * 128 + lane * 4;
    atomicAdd(base + 0, hv.x * norm);
    atomicAdd(base + 1, hv.y * norm);
    atomicAdd(base + 2, hv.z * norm);
    atomicAdd(base + 3, hv.w * norm);
}

// ---------------- edge scatter, 64 channels: wave per edge, float2/lane ----------------
__global__ void gcn_edge_scatter64(const int* __restrict__ src, const int* __restrict__ dst,
                                   const float* __restrict__ ew, const float* __restrict__ dinv,
                                   const float* __restrict__ H, float* __restrict__ Agg,
                                   int n_edges) {
    int e = blockIdx.x * (blockDim.x >> 5) + (threadIdx.x >> 5);
    if (e >= n_edges) return;
    int lane = threadIdx.x & 31;
    int s = src[e], d = dst[e];
    float norm = dinv[s] * ew[e] * dinv[d];
    const float2 hv = *(const float2*)(H + (size_t)s * 64 + lane * 2);
    float* base = Agg + (size_t)d * 64 + lane * 2;
    atomicAdd(base + 0, hv.x * norm);
    atomicAdd(base + 1, hv.y * norm);
}

// ---------------- in-place ReLU (float4) ----------------
__global__ void gcn_relu4(float* __restrict__ p, int n4) {
    int i = blockIdx.x * blockDim.x + threadIdx.x;
    if (i < n4) {
        float4 v = *(float4*)(p + 4 * (size_t)i);
        v.x = fmaxf(v.x, 0.f); v.y = fmaxf(v.y, 0.f);
        v.z = fmaxf(v.z, 0.f); v.w = fmaxf(v.w, 0.f);
        *(float4*)(p + 4 * (size_t)i) = v;
    }
}

extern "C" void kernel_launch(void* const* d_in, const int* in_sizes, int n_in,
                              void* d_out, int out_size, void* d_ws, size_t ws_size,
                              hipStream_t stream) {
    const float* x  = (const float*)d_in[0];
    const int*  eix = (const int*)d_in[1];
    const float* ew = (const float*)d_in[2];
    const float* W1 = (const float*)d_in[3];
    const float* b1 = (const float*)d_in[4];
    const float* W2 = (const float*)d_in[5];
    const float* b2 = (const float*)d_in[6];

    const int n_nodes = in_sizes[0] / IN_C;
    const int n_edges = in_sizes[2];
    const int hid_c = 128, out_c = 64;
    const int* src = eix;
    const int* dst = eix + n_edges;

    // Workspace layout (floats): dinv | h1[n,128] | agg1[n,128] | h2[n,64]
    float* dinv = (float*)d_ws;
    size_t off = ((size_t)n_nodes + 3) & ~(size_t)3;    // keep 16B alignment
    float* h1   = dinv + off;
    float* agg1 = h1 + (size_t)n_nodes * hid_c;
    float* h2   = agg1 + (size_t)n_nodes * hid_c;
    float* out  = (float*)d_out;

    const int B = 256;
    // 1) degrees -> dinv (in place)
    gcn_deg_init<<<(n_nodes + B - 1) / B, B, 0, stream>>>(dinv, n_nodes);
    gcn_deg_accum<<<(n_edges + B - 1) / B, B, 0, stream>>>(dst, ew, dinv, n_edges);
    gcn_deg_to_dinv<<<(n_nodes + B - 1) / B, B, 0, stream>>>(dinv, n_nodes);

    // 2) h1 = x@W1 ; agg1 = h1*dinv^2 + b1   (WMMA, 8 waves/block)
    gcn_gemm_wmma<<<(n_nodes + 15) / 16, 2 * hid_c, 0, stream>>>(
        x, W1, b1, dinv, h1, agg1, n_nodes, hid_c);

    // 3) agg1 += scatter(norm * h1[src])
    gcn_edge_scatter128<<<(n_edges + 7) / 8, B, 0, stream>>>(
        src, dst, ew, dinv, h1, agg1, n_edges);

    // 4) ReLU in place
    int n4 = (n_nodes * hid_c) / 4;
    gcn_relu4<<<(n4 + B - 1) / B, B, 0, stream>>>(agg1, n4);

    // 5) h2 = relu(agg1)@W2 ; out = h2*dinv^2 + b2   (WMMA, 4 waves/block)
    gcn_gemm_wmma<<<(n_nodes + 15) / 16, 2 * out_c, 0, stream>>>(
        agg1, W2, b2, dinv, h2, out, n_nodes, out_c);

    // 6) out += scatter(norm * h2[src])
    gcn_edge_scatter64<<<(n_edges + 7) / 8, B, 0, stream>>>(
        src, dst, ew, dinv, h2, out, n_edges);
}